// DrugGraphGNN_2224793059951
// MI455X (gfx1250) — compile-verified
//
#include <hip/hip_runtime.h>
#include <hip/hip_bf16.h>

// DrugGraphGNN for MI455X (gfx1250, wave32).
// Pipeline: deg/dinv -> [GEMM(WMMA f32) -> zero -> edge scatter-add -> fused
// selfloop+bias+ReLU] x2 -> mean pool -> GEMM(WMMA f32)+bias.
// fp32 WMMA (16x16x4) chosen: GEMM is ~3.3 GFLOP total (noise); the L2-resident
// edge scatter (~1.1 GB L2 traffic, ~140M f32 atomics) dominates, so we keep
// reference precision for free.

#define NN 50000   // nodes
#define NE 500000  // edges
#define DD 128     // feature dim (in = hid = out)
#define NG 1024    // graphs

typedef float v2f __attribute__((ext_vector_type(2)));
typedef float v8f __attribute__((ext_vector_type(8)));

// deg[i] = 1 (self loop) + in-degree at targets
__global__ void k_init_deg(float* __restrict__ deg) {
  int i = blockIdx.x * blockDim.x + threadIdx.x;
  if (i < NN) deg[i] = 1.0f;
}

__global__ void k_count_deg(const int* __restrict__ col, float* __restrict__ deg) {
  int e = blockIdx.x * blockDim.x + threadIdx.x;
  if (e < NE) atomicAdd(&deg[col[e]], 1.0f);
}

__global__ void k_dinv(const float* __restrict__ deg, float* __restrict__ dinv) {
  int i = blockIdx.x * blockDim.x + threadIdx.x;
  if (i < NN) dinv[i] = 1.0f / sqrtf(deg[i]);  // deg >= 1 always (self loops)
}

// C[M,128] = A[M,128] @ B[128,128] (+ bias). One wave per 16x16 C tile,
// block = 8 waves covering all 8 N-tiles of one 16-row M-tile.
// K accumulated with v_wmma_f32_16x16x4_f32 (32 steps).
__global__ __launch_bounds__(256) void k_gemm_wmma(
    const float* __restrict__ A, const float* __restrict__ B,
    const float* __restrict__ bias, float* __restrict__ C, int M) {
  const int lane = threadIdx.x & 31;
  const int wave = threadIdx.x >> 5;
  const int m0   = blockIdx.x << 4;
  const int n0   = wave << 4;
  const int half = lane >> 4;   // 0: K pair {k0,k0+1}, 1: {k0+2,k0+3}
  const int l    = lane & 15;   // A: row within tile; B/C: column within tile

  int arow = m0 + l;
  if (arow >= M) arow = M - 1;  // clamp (stores are masked); no EXEC divergence
  const float* __restrict__ Ap = A + (size_t)arow * DD;

  v8f acc = {};
#pragma unroll
  for (int k0 = 0; k0 < DD; k0 += 4) {
    const int kk = k0 + half * 2;
    v2f a, b;
    a.x = Ap[kk];
    a.y = Ap[kk + 1];
    b.x = B[kk * DD + n0 + l];
    b.y = B[(kk + 1) * DD + n0 + l];
    acc = __builtin_amdgcn_wmma_f32_16x16x4_f32(
        /*neg_a=*/false, a, /*neg_b=*/false, b,
        /*c_mod=*/(short)0, acc, /*reuse_a=*/false, /*reuse_b=*/false);
  }

  const float bj = bias ? bias[n0 + l] : 0.0f;
#pragma unroll
  for (int v = 0; v < 8; ++v) {
    const int r = m0 + v + half * 8;  // C/D layout: VGPR v -> row v (+8 upper half)
    if (r < M) C[(size_t)r * DD + n0 + l] = acc[v] + bj;
  }
}

// One wave per edge: lane moves 4 contiguous floats (float4 gather at src,
// 4x global_atomic_add_f32 at dst). h (25.6MB) is L2-resident.
__global__ __launch_bounds__(256) void k_edge_scatter(
    const float* __restrict__ h, const int* __restrict__ row,
    const int* __restrict__ col, const float* __restrict__ dinv,
    float* __restrict__ acc) {
  const int wid  = (blockIdx.x * blockDim.x + threadIdx.x) >> 5;
  const int lane = threadIdx.x & 31;
  if (wid >= NE) return;
  const int r = row[wid];
  const int c = col[wid];
  const float norm = dinv[r] * dinv[c];
  const float4 m = ((const float4*)(h + (size_t)r * DD))[lane];
  float* o = acc + (size_t)c * DD + lane * 4;
  atomicAdd(o + 0, m.x * norm);
  atomicAdd(o + 1, m.y * norm);
  atomicAdd(o + 2, m.z * norm);
  atomicAdd(o + 3, m.w * norm);
}

// acc[i] = relu(acc[i] + h[i]*dinv(node)^2 + bias[j])   (self loop fused; in place)
__global__ void k_finalize_relu(float* __restrict__ acc, const float* __restrict__ h,
                                const float* __restrict__ dinv,
                                const float* __restrict__ bias) {
  const int i = blockIdx.x * blockDim.x + threadIdx.x;
  if (i >= NN * DD) return;
  const int node = i >> 7;
  const int j = i & (DD - 1);
  const float di = dinv[node];
  const float v = acc[i] + h[i] * di * di + bias[j];
  acc[i] = fmaxf(v, 0.0f);
}

// One wave per node: accumulate node features into its graph's sum.
__global__ __launch_bounds__(256) void k_pool_sum(
    const float* __restrict__ x, const int* __restrict__ batch,
    float* __restrict__ sums, float* __restrict__ cnts) {
  const int wid  = (blockIdx.x * blockDim.x + threadIdx.x) >> 5;
  const int lane = threadIdx.x & 31;
  if (wid >= NN) return;
  const int g = batch[wid];
  const float4 m = ((const float4*)(x + (size_t)wid * DD))[lane];
  float* o = sums + (size_t)g * DD + lane * 4;
  atomicAdd(o + 0, m.x);
  atomicAdd(o + 1, m.y);
  atomicAdd(o + 2, m.z);
  atomicAdd(o + 3, m.w);
  if (lane == 0) atomicAdd(&cnts[g], 1.0f);
}

__global__ void k_pool_div(const float* __restrict__ sums,
                           const float* __restrict__ cnts,
                           float* __restrict__ pooled) {
  const int i = blockIdx.x * blockDim.x + threadIdx.x;
  if (i < NG * DD) pooled[i] = sums[i] / fmaxf(cnts[i >> 7], 1.0f);
}

extern "C" void kernel_launch(void* const* d_in, const int* in_sizes, int n_in,
                              void* d_out, int out_size, void* d_ws, size_t ws_size,
                              hipStream_t stream) {
  (void)in_sizes; (void)n_in; (void)out_size; (void)ws_size;
  const float* x   = (const float*)d_in[0];
  const int*   ei  = (const int*)d_in[1];   // [2, NE]
  const int*   bat = (const int*)d_in[2];   // [NN]
  const float* W1  = (const float*)d_in[3];
  const float* b1  = (const float*)d_in[4];
  const float* W2  = (const float*)d_in[5];
  const float* b2  = (const float*)d_in[6];
  const float* Wf  = (const float*)d_in[7];
  const float* bf  = (const float*)d_in[8];
  float* out = (float*)d_out;
  const int* row = ei;        // sources
  const int* col = ei + NE;   // targets

  // Workspace layout (256B aligned): deg, dinv, bufA (h), bufB (acc/x), sums,
  // cnts, pooled. Total ~52.6 MB.
  auto up = [](size_t v) { return (v + 255) & ~(size_t)255; };
  char* w = (char*)d_ws;
  size_t o = 0;
  float* deg    = (float*)(w + o); o = up(o + (size_t)NN * 4);
  float* dinv   = (float*)(w + o); o = up(o + (size_t)NN * 4);
  float* bufA   = (float*)(w + o); o = up(o + (size_t)NN * DD * 4);
  float* bufB   = (float*)(w + o); o = up(o + (size_t)NN * DD * 4);
  float* sums   = (float*)(w + o); o = up(o + (size_t)NG * DD * 4);
  float* cnts   = (float*)(w + o); o = up(o + (size_t)NG * 4);
  float* pooled = (float*)(w + o); o = up(o + (size_t)NG * DD * 4);

  const int B256 = 256;
  const size_t featBytes = (size_t)NN * DD * 4;

  // --- degrees / symmetric norm ---
  k_init_deg<<<(NN + B256 - 1) / B256, B256, 0, stream>>>(deg);
  k_count_deg<<<(NE + B256 - 1) / B256, B256, 0, stream>>>(col, deg);
  k_dinv<<<(NN + B256 - 1) / B256, B256, 0, stream>>>(deg, dinv);

  // --- layer 1: h = x@W1 ; acc = scatter ; B = relu(acc + selfloop + b1) ---
  k_gemm_wmma<<<NN / 16, B256, 0, stream>>>(x, W1, nullptr, bufA, NN);
  hipMemsetAsync(bufB, 0, featBytes, stream);
  k_edge_scatter<<<(NE * 32) / B256, B256, 0, stream>>>(bufA, row, col, dinv, bufB);
  k_finalize_relu<<<(NN * DD) / B256, B256, 0, stream>>>(bufB, bufA, dinv, b1);

  // --- layer 2: h = B@W2 ; acc = scatter ; B = relu(acc + selfloop + b2) ---
  k_gemm_wmma<<<NN / 16, B256, 0, stream>>>(bufB, W2, nullptr, bufA, NN);
  hipMemsetAsync(bufB, 0, featBytes, stream);
  k_edge_scatter<<<(NE * 32) / B256, B256, 0, stream>>>(bufA, row, col, dinv, bufB);
  k_finalize_relu<<<(NN * DD) / B256, B256, 0, stream>>>(bufB, bufA, dinv, b2);

  // --- global mean pool ---
  hipMemsetAsync(sums, 0, (size_t)NG * DD * 4, stream);
  hipMemsetAsync(cnts, 0, (size_t)NG * 4, stream);
  k_pool_sum<<<(NN * 32 + B256 - 1) / B256, B256, 0, stream>>>(bufB, bat, sums, cnts);
  k_pool_div<<<(NG * DD) / B256, B256, 0, stream>>>(sums, cnts, pooled);

  // --- readout: out = pooled @ Wf + bf ---
  k_gemm_wmma<<<NG / 16, B256, 0, stream>>>(pooled, Wf, bf, out, NG);
}